// SpeechEncDec_34102040330494
// MI455X (gfx1250) — compile-verified
//
#include <hip/hip_runtime.h>
#include <hip/hip_bf16.h>
#include <math.h>

// ---------------------------------------------------------------------------
// SpeechEncDec for MI455X (gfx1250, wave32).
//
// Roofline: 805 MB of fp32 GRU weights @ 23.3 TB/s => ~35us HBM floor;
// 103 GFLOP of matmul is negligible, so we stay fp32 end-to-end and use
// V_WMMA_F32_16X16X4_F32 (no conversion traffic, full precision through 512
// tanh/sigmoid layers).
//
// The critical path is the 2 x 256-layer x 64-step recurrence. We WAVEFRONT-
// PIPELINE the stack: 64 layers launched together (one WG each); layer l
// spins on a per-step flag from layer l-1 (acquire/release atomics +
// s_sleep). A 64-layer chunk drains in ~(64+64) pipeline steps with all
// WGPs streaming weights from L2 concurrently. gx is folded into the
// per-step GEMM: each step computes x_t*Wih^T and h*Whh^T = 96 WMMA 16x16
// tiles. HBM traffic is unchanged (weights read once, L2-resident/layer).
//
// Round-4 change: round 3's register FIFO at 1024 threads/block spilled to
// scratch (tiny per-lane VGPR budget at 32 waves/WGP). The recurrence block
// is now 256 threads (8 waves, 12 tiles each) with the simple inner loop the
// scheduler already pipelines ~20 loads deep at this occupancy; the GPU
// packs several layer-blocks per WGP (56 KB LDS each), so WGP-level latency
// hiding is preserved while per-wave registers quadruple.
// ---------------------------------------------------------------------------

#define H       256
#define THREE_H 768
#define NL      256
#define TT      64
#define BB      4
#define NCLS    26
#define CHUNK   64          // layers pipelined per launch (co-resident WGs)
#define RBLK    256         // recurrence block size (8 waves)

typedef __attribute__((ext_vector_type(2))) float v2f;
typedef __attribute__((ext_vector_type(8))) float v8f;

__device__ __forceinline__ float sigmoidf_(float x) { return 1.0f / (1.0f + __expf(-x)); }

// ---------------------------------------------------------------------------
// Front end: conv1 (1x1, 1->128) + conv2 (1x1, 128->256) + stride-2 subsample
// + transpose to seq[t][b][h]. 8.4 MMAC total -> plain VALU.
// ---------------------------------------------------------------------------
__global__ __launch_bounds__(256) void sed_front_kernel(
    const float* __restrict__ X,        // [B,1,128]
    const float* __restrict__ w1,       // [128,1]
    const float* __restrict__ b1,       // [128]
    const float* __restrict__ w2,       // [256,128]
    const float* __restrict__ b2,       // [256]
    float* __restrict__ seq)            // [T,B,H]
{
    int idx = blockIdx.x * 256 + threadIdx.x;  // 65536 = T*B*H
    int o   = idx & 255;
    int row = idx >> 8;                        // t*B + b
    int t   = row >> 2;
    int b   = row & 3;
    float xv = X[b * 128 + 2 * t];
    float acc = b2[o];
    const float* w2o = w2 + o * 128;
    for (int c = 0; c < 128; ++c)
        acc += (xv * w1[c] + b1[c]) * w2o[c];
    seq[idx] = acc;
}

// ---------------------------------------------------------------------------
// Pipelined GRU chunk: blockIdx.x = layer-in-chunk (0..63). Per step t:
//   wait(flags[li-1] >= t+1)  -> load x_t into padded LDS A tile
//   96 WMMA tiles (48 gx = x*Wih^T + bih, 48 gh = h*Whh^T + bhh), M pad 4->16
//   gate math + h update (4 (b,j) pairs per thread), write row t, signal.
// WMMA f32 16x16x4 lane maps: A/B: lane<16 K={0,1}, lane>=16 K={2,3};
// C/D: VGPR i = row i (+8 for hi lanes), col = lane&15.
// ---------------------------------------------------------------------------
__global__ __launch_bounds__(RBLK) void sed_rec_pipe_kernel(
    const float* __restrict__ in_base,   // input sequence for chunk-layer 0
    float* __restrict__ bufs,            // CHUNK x [T*B*H] inter-layer buffers
    float* __restrict__ last_out,        // output buffer for chunk-layer 63
    const float* __restrict__ Wih_base,  // [CHUNK,3H,H]
    const float* __restrict__ Whh_base,  // [CHUNK,3H,H]
    const float* __restrict__ bih_base,  // [CHUNK,3H]
    const float* __restrict__ bhh_base,  // [CHUNK,3H]
    int* flags,                          // [CHUNK] progress counters (pre-zeroed)
    int relu_last)                       // ReLU on chunk-layer 63 output
{
    __shared__ float a_lds[2 * 16 * H];       // [0]=x_t (padded 16 rows), [1]=h
    __shared__ float g_lds[2 * BB * THREE_H]; // [0]=gx rows 0..3, [1]=gh

    const int li  = blockIdx.x;
    const int tid = threadIdx.x;
    const size_t SEQ = (size_t)TT * BB * H;

    const float* in  = (li == 0) ? in_base : (bufs + (size_t)(li - 1) * SEQ);
    float*       out = (li == CHUNK - 1) ? last_out : (bufs + (size_t)li * SEQ);
    const float* Wih = Wih_base + (size_t)li * THREE_H * H;
    const float* Whh = Whh_base + (size_t)li * THREE_H * H;
    const float* bih = bih_base + (size_t)li * THREE_H;
    const float* bhh = bhh_base + (size_t)li * THREE_H;
    const int do_relu = relu_last && (li == CHUNK - 1);

    for (int i = tid; i < 2 * 16 * H; i += RBLK) a_lds[i] = 0.0f;  // h0=0 + M-pad
    __syncthreads();

    const int wave   = tid >> 5;              // 0..7
    const int lane   = tid & 31;
    const int lane16 = lane & 15;
    const int khi    = (lane >> 4) << 1;      // 0 or 2

    for (int t = 0; t < TT; ++t) {
        // ---- consume: wait for layer li-1 to have produced step t
        if (li > 0 && tid == 0) {
            while (__hip_atomic_load(flags + (li - 1), __ATOMIC_ACQUIRE,
                                     __HIP_MEMORY_SCOPE_AGENT) < t + 1)
                __builtin_amdgcn_s_sleep(1);
        }
        __syncthreads();
        // x_t rows 0..3 ([b][j] layout matches the input row's 1024 floats)
        for (int i = tid; i < BB * H; i += RBLK)
            a_lds[i] = in[(size_t)t * BB * H + i];
        __syncthreads();

        // ---- 96 WMMA tiles over 8 waves: tt<48 -> gx GEMM, tt>=48 -> gh GEMM
        for (int s = 0; s < 12; ++s) {
            int tt    = s * 8 + wave;                  // 0..95, wave-uniform
            int is_gh = (tt >= 48) ? 1 : 0;
            int gt    = tt - (is_gh ? 48 : 0);         // column tile 0..47
            const float* arow  = a_lds + is_gh * (16 * H) + lane16 * H + khi;
            const float* wbase = is_gh ? Whh : Wih;
            const float* bvec  = is_gh ? bhh : bih;
            const float* brow  = wbase + (size_t)(gt * 16 + lane16) * H + khi;
            float bias = bvec[gt * 16 + lane16];
            v8f c = { bias, bias, bias, bias, bias, bias, bias, bias };
            for (int k0 = 0; k0 < H; k0 += 4) {
                v2f a = { arow[k0], arow[k0 + 1] };
                v2f b = { brow[k0], brow[k0 + 1] };
                c = __builtin_amdgcn_wmma_f32_16x16x4_f32(false, a, false, b,
                                                          (short)0, c, false, false);
            }
            // batch rows 0..3 live in VGPRs 0..3 on lanes 0..15
            if (lane < 16) {
                float* drow = g_lds + is_gh * (BB * THREE_H) + gt * 16 + lane;
#pragma unroll
                for (int i = 0; i < BB; ++i)
                    drow[i * THREE_H] = c[i];
            }
        }
        __syncthreads();

        // ---- gates + state update (4 (b,j) pairs per thread, j = tid)
        {
            int j = tid;  // 0..255
#pragma unroll
            for (int b = 0; b < BB; ++b) {
                const float* gxr_p = g_lds + b * THREE_H;
                const float* ghr_p = g_lds + BB * THREE_H + b * THREE_H;
                float r = sigmoidf_(gxr_p[j] + ghr_p[j]);
                float z = sigmoidf_(gxr_p[H + j] + ghr_p[H + j]);
                float n = tanhf(gxr_p[2 * H + j] + r * ghr_p[2 * H + j]);
                float hp = a_lds[16 * H + b * H + j];
                float hn = (1.0f - z) * n + z * hp;
                a_lds[16 * H + b * H + j] = hn;
                out[(size_t)(t * BB + b) * H + j] = do_relu ? fmaxf(hn, 0.0f) : hn;
            }
        }
        __threadfence();
        __syncthreads();

        // ---- produce: publish step t for layer li+1
        if (tid == 0)
            __hip_atomic_store(flags + li, t + 1, __ATOMIC_RELEASE,
                               __HIP_MEMORY_SCOPE_AGENT);
    }
}

// ---------------------------------------------------------------------------
// Attention + classifier. One block per (t,b). Reference's quirky indexing:
// c[t,b] = attn[t,b,0]*emb[0,b] + sum_{s<T-1} attn[t,b,s]*emb[s+1,b].
// ---------------------------------------------------------------------------
__global__ __launch_bounds__(256) void sed_attn_kernel(
    const float* __restrict__ hid,      // [T,B,H]
    const float* __restrict__ emb,      // [T,B,H]
    const float* __restrict__ fclw,     // [NCLS, 2H]
    const float* __restrict__ fclb,     // [NCLS]
    float* __restrict__ out)            // [T,B,NCLS]
{
    __shared__ float sc[TT];
    __shared__ float av[TT];
    __shared__ float cbuf[H];
    __shared__ float lg[NCLS];

    int t = blockIdx.x >> 2;
    int b = blockIdx.x & 3;
    int tid = threadIdx.x;
    const float* hrow = hid + (size_t)(t * BB + b) * H;

    if (tid < TT) {
        const float* erow = emb + (size_t)(tid * BB + b) * H;
        float acc = 0.0f;
        for (int d = 0; d < H; ++d) acc += hrow[d] * erow[d];
        sc[tid] = acc;
    }
    __syncthreads();
    if (tid == 0) {
        float m = sc[0];
        for (int s = 1; s < TT; ++s) m = fmaxf(m, sc[s]);
        float sum = 0.0f;
        for (int s = 0; s < TT; ++s) { float e = __expf(sc[s] - m); av[s] = e; sum += e; }
        float inv = 1.0f / sum;
        for (int s = 0; s < TT; ++s) av[s] *= inv;
    }
    __syncthreads();
    {
        int d = tid;  // 0..255
        float cv = av[0] * emb[(size_t)b * H + d];
        for (int s = 0; s < TT - 1; ++s)
            cv += av[s] * emb[(size_t)((s + 1) * BB + b) * H + d];
        cbuf[d] = cv;
    }
    __syncthreads();
    if (tid < NCLS) {
        const float* w = fclw + (size_t)tid * (2 * H);
        float acc = fclb[tid];
        for (int d = 0; d < H; ++d) acc += hrow[d] * w[d];
        for (int d = 0; d < H; ++d) acc += cbuf[d] * w[H + d];
        lg[tid] = acc;
    }
    __syncthreads();
    if (tid == 0) {
        float m = lg[0];
        for (int o = 1; o < NCLS; ++o) m = fmaxf(m, lg[o]);
        float ex[NCLS]; float sum = 0.0f;
        for (int o = 0; o < NCLS; ++o) { ex[o] = __expf(lg[o] - m); sum += ex[o]; }
        float inv = 1.0f / sum;
        for (int o = 0; o < NCLS; ++o)
            out[(size_t)(t * BB + b) * NCLS + o] = ex[o] * inv;
    }
}

// ---------------------------------------------------------------------------
// Host driver. Workspace (floats):
//   bufs   : CHUNK * T*B*H  = 4,194,304   (16 MB inter-layer pipeline buffers)
//   seq0   : 65,536     bndA/bndB: 2*65,536   emb: 65,536   hid: 65,536
//   flags  : CHUNK ints
// Total ~17.3 MB. 18 stream ops, fully deterministic, graph-capturable
// (hipMemsetAsync only).
// ---------------------------------------------------------------------------
extern "C" void kernel_launch(void* const* d_in, const int* in_sizes, int n_in,
                              void* d_out, int out_size, void* d_ws, size_t ws_size,
                              hipStream_t stream)
{
    const float* X       = (const float*)d_in[0];
    const float* conv1_w = (const float*)d_in[1];
    const float* conv1_b = (const float*)d_in[2];
    const float* conv2_w = (const float*)d_in[3];
    const float* conv2_b = (const float*)d_in[4];
    const float* enc_Wih = (const float*)d_in[5];
    const float* enc_Whh = (const float*)d_in[6];
    const float* enc_bih = (const float*)d_in[7];
    const float* enc_bhh = (const float*)d_in[8];
    const float* dec_Wih = (const float*)d_in[9];
    const float* dec_Whh = (const float*)d_in[10];
    const float* dec_bih = (const float*)d_in[11];
    const float* dec_bhh = (const float*)d_in[12];
    const float* fcl_w   = (const float*)d_in[13];
    const float* fcl_b   = (const float*)d_in[14];

    const size_t SEQ = (size_t)TT * BB * H;     // 65536
    float* ws   = (float*)d_ws;
    float* bufs = ws;                            // CHUNK * SEQ
    float* seq0 = bufs + (size_t)CHUNK * SEQ;
    float* bndA = seq0 + SEQ;
    float* bndB = bndA + SEQ;
    float* emb  = bndB + SEQ;
    float* hid  = emb  + SEQ;
    int*   flags = (int*)(hid + SEQ);

    const size_t WSTRIDE = (size_t)CHUNK * THREE_H * H;   // weights per chunk
    const size_t BSTRIDE = (size_t)CHUNK * THREE_H;       // biases per chunk

    sed_front_kernel<<<256, 256, 0, stream>>>(X, conv1_w, conv1_b, conv2_w, conv2_b, seq0);

    // Encoder: 4 pipelined chunks of 64 layers; ReLU folded into last output.
    const float* chunk_in = seq0;
    for (int c = 0; c < 4; ++c) {
        hipMemsetAsync(flags, 0, CHUNK * sizeof(int), stream);
        float* last = (c == 3) ? emb : ((c & 1) ? bndB : bndA);
        sed_rec_pipe_kernel<<<CHUNK, RBLK, 0, stream>>>(
            chunk_in, bufs, last,
            enc_Wih + (size_t)c * WSTRIDE, enc_Whh + (size_t)c * WSTRIDE,
            enc_bih + (size_t)c * BSTRIDE, enc_bhh + (size_t)c * BSTRIDE,
            flags, (c == 3) ? 1 : 0);
        chunk_in = last;
    }

    // Decoder: 4 pipelined chunks of 64 layers.
    chunk_in = emb;
    for (int c = 0; c < 4; ++c) {
        hipMemsetAsync(flags, 0, CHUNK * sizeof(int), stream);
        float* last = (c == 3) ? hid : ((c & 1) ? bndB : bndA);
        sed_rec_pipe_kernel<<<CHUNK, RBLK, 0, stream>>>(
            chunk_in, bufs, last,
            dec_Wih + (size_t)c * WSTRIDE, dec_Whh + (size_t)c * WSTRIDE,
            dec_bih + (size_t)c * BSTRIDE, dec_bhh + (size_t)c * BSTRIDE,
            flags, 0);
        chunk_in = last;
    }

    sed_attn_kernel<<<TT * BB, 256, 0, stream>>>(hid, emb, fcl_w, fcl_b, (float*)d_out);
}